// SparseBlock_87282325389633
// MI455X (gfx1250) — compile-verified
//
#include <hip/hip_runtime.h>
#include <hip/hip_bf16.h>

typedef __attribute__((ext_vector_type(2))) float v2f;
typedef __attribute__((ext_vector_type(4))) float v4f;
typedef __attribute__((ext_vector_type(8))) float v8f;

#define NKERN 27
#define COUT  32
#define EPSBN 1e-5f

__device__ __forceinline__ v8f wmma_f32(v2f a, v2f b, v8f c) {
  // V_WMMA_F32_16X16X4_F32 : D = A(16x4) * B(4x16) + C(16x16), all fp32
  return __builtin_amdgcn_wmma_f32_16x16x4_f32(
      /*neg_a=*/false, a, /*neg_b=*/false, b,
      /*c_mod=*/(short)0, c, /*reuse_a=*/false, /*reuse_b=*/false);
}

// Repack W [K][Cin][32] into WMMA B-fragment order:
// dst[(k*nchunk+ch)*32 + lane] = float4{ B0.x, B0.y, B1.x, B1.y }
// where for lane t (n=t%16, half=t/16, cc=ch*4+2*half):
//   B0 covers cout 0..15, B1 covers cout 16..31, K-pair (cc, cc+1).
__global__ __launch_bounds__(256) void repack_kernel(
    const float* __restrict__ W, float* __restrict__ dst, int Kk, int Cin, int nchunk) {
  int tid = blockIdx.x * blockDim.x + threadIdx.x;
  int total = Kk * nchunk * 32;
  if (tid >= total) return;
  int t  = tid & 31;
  int kc = tid >> 5;                 // k*nchunk + ch
  int ch = kc % nchunk;
  int n = t & 15, half = t >> 4;
  int k = kc / nchunk;
  int cc = ch * 4 + 2 * half;
  const float* wrow = W + ((size_t)k * Cin + cc) * COUT;   // row stride COUT
  v4f v;
  v.x = wrow[n];             // B[cc  ][n]
  v.y = wrow[COUT + n];      // B[cc+1][n]
  v.z = wrow[16 + n];        // B[cc  ][16+n]
  v.w = wrow[COUT + 16 + n]; // B[cc+1][16+n]
  ((v4f*)dst)[(size_t)kc * 32 + t] = v;
}

__global__ __launch_bounds__(256) void zero_kernel(float* p, int n) {
  int i = blockIdx.x * blockDim.x + threadIdx.x;
  if (i < n) p[i] = 0.0f;
}

// One wave per 16-voxel tile. Gathered GEMM: acc[16x32] += A_gather[16 x 4] * Bfrag[4 x 32]
// over k in [0,27), chunk in [0,NCHUNK). Epilogue: affine (bias+BN folded), optional
// residual + ReLU (stage2) or ReLU into h (stage1).
template <int NCHUNK, bool STAGE2>
__global__ __launch_bounds__(256) void conv_stage(
    const float* __restrict__ src,     // gathered source rows [Nv][NCHUNK*4]
    const int*   __restrict__ nbr,     // [Nv][27], value Nv => zero row
    const float* __restrict__ bfrag,   // repacked weights, v4f[27*NCHUNK][32]
    const float* __restrict__ bias,
    const float* __restrict__ gamma,
    const float* __restrict__ beta,
    const float* __restrict__ mean,
    const float* __restrict__ var,
    const float* __restrict__ zerorow,
    const float* __restrict__ resid,   // feats (stage2) or nullptr
    float* __restrict__ dst,
    int Nv, int colBase) {
  const int CIN = NCHUNK * 4;
  int wave   = (blockIdx.x * blockDim.x + threadIdx.x) >> 5;
  int nTiles = (Nv + 15) >> 4;
  if (wave >= nTiles) return;                 // wave-uniform: EXEC stays all-1s below

  int t = threadIdx.x & 31;
  int m = t & 15, half = t >> 4;
  int i0 = wave << 4;
  int gr  = i0 + m;
  int grc = gr < Nv ? gr : Nv - 1;

  v8f acc0 = {};
  v8f acc1 = {};

  const v4f* __restrict__ bbase = ((const v4f*)bfrag) + t;

#pragma unroll 1
  for (int k = 0; k < NKERN; ++k) {
    int idx = nbr[(size_t)grc * NKERN + k];
    const float* arow = (idx < Nv) ? (src + (size_t)idx * CIN) : zerorow;
    const float* ar = arow + 2 * half;        // this lane's K-pair within each chunk
    const v4f* bp = bbase + (size_t)(k * NCHUNK) * 32;
#pragma unroll
    for (int ch = 0; ch < NCHUNK; ++ch) {
      v2f a = *(const v2f*)(ar + ch * 4);     // contiguous float2 gather
      v4f b = bp[(size_t)ch * 32];            // both cout tiles in one b128
      v2f b0; b0.x = b.x; b0.y = b.y;
      v2f b1; b1.x = b.z; b1.y = b.w;
      acc0 = wmma_f32(a, b0, acc0);
      acc1 = wmma_f32(a, b1, acc1);
    }
  }

  // Folded bias + BN affine: out = acc*s + sh, s = g*rsqrt(var+eps), sh = (b-mean)*s + beta
  float s0  = gamma[m]      * rsqrtf(var[m]      + EPSBN);
  float sh0 = (bias[m]      - mean[m])      * s0 + beta[m];
  float s1  = gamma[16 + m] * rsqrtf(var[16 + m] + EPSBN);
  float sh1 = (bias[16 + m] - mean[16 + m]) * s1 + beta[16 + m];

#pragma unroll
  for (int r = 0; r < 8; ++r) {
    int row = i0 + r + 8 * half;              // C/D layout: vgpr r, halves are M and M+8
    if (row < Nv) {
      float o0 = acc0[r] * s0 + sh0;
      float o1 = acc1[r] * s1 + sh1;
      if (STAGE2) {
        size_t base = (size_t)row * 96 + colBase;
        o0 = fmaxf(o0 + resid[base + m],      0.0f);
        o1 = fmaxf(o1 + resid[base + 16 + m], 0.0f);
        dst[base + m]      = o0;
        dst[base + 16 + m] = o1;
      } else {
        size_t base = (size_t)row * 32;
        dst[base + m]      = fmaxf(o0, 0.0f); // ReLU after bn1
        dst[base + 16 + m] = fmaxf(o1, 0.0f);
      }
    }
  }
}

extern "C" void kernel_launch(void* const* d_in, const int* in_sizes, int n_in,
                              void* d_out, int out_size, void* d_ws, size_t ws_size,
                              hipStream_t stream) {
  const float* feats = (const float*)d_in[0];
  const int* nbr[3] = {(const int*)d_in[1], (const int*)d_in[2], (const int*)d_in[3]};
  int Nv = in_sizes[0] / 96;

  // Param leaves start at index 4, 12 per branch. Two plausible flattenings:
  //  sorted keys   : W1, W2, b1, b2, bn1{beta,gamma,mean,var}, bn2{beta,gamma,mean,var}
  //  insertion keys: W1, b1, bn1{gamma,beta,mean,var}, W2, b2, bn2{gamma,beta,mean,var}
  struct Br { const float *W1,*b1,*g1,*be1,*m1,*v1,*W2,*b2,*g2,*be2,*m2,*v2; } br[3];
  bool sorted_keys = (n_in > 5) && (in_sizes[5] == NKERN * COUT * COUT);
  int p = 4;
  for (int b = 0; b < 3; ++b) {
    const float* q[12];
    for (int j = 0; j < 12; ++j) q[j] = (const float*)d_in[p + j];
    if (sorted_keys) {
      br[b].W1 = q[0];  br[b].W2 = q[1];
      br[b].b1 = q[2];  br[b].b2 = q[3];
      br[b].be1 = q[4]; br[b].g1 = q[5]; br[b].m1 = q[6];  br[b].v1 = q[7];
      br[b].be2 = q[8]; br[b].g2 = q[9]; br[b].m2 = q[10]; br[b].v2 = q[11];
    } else {
      br[b].W1 = q[0];  br[b].b1 = q[1];
      br[b].g1 = q[2];  br[b].be1 = q[3]; br[b].m1 = q[4];  br[b].v1 = q[5];
      br[b].W2 = q[6];  br[b].b2 = q[7];
      br[b].g2 = q[8];  br[b].be2 = q[9]; br[b].m2 = q[10]; br[b].v2 = q[11];
    }
    p += 12;
  }

  // Workspace layout (floats): zero row | 3x W1 frags | 3x W2 frags | h buffer
  const size_t w1frag_sz = (size_t)NKERN * 24 * 32 * 4;  // 82944
  const size_t w2frag_sz = (size_t)NKERN * 8 * 32 * 4;   // 27648
  float* ws = (float*)d_ws;
  float* zerorow = ws;
  size_t off = 128;
  float* b1f[3]; float* b2f[3];
  for (int b = 0; b < 3; ++b) { b1f[b] = ws + off; off += w1frag_sz; }
  for (int b = 0; b < 3; ++b) { b2f[b] = ws + off; off += w2frag_sz; }
  float* h = ws + off;

  zero_kernel<<<1, 128, 0, stream>>>(zerorow, 128);
  for (int b = 0; b < 3; ++b) {
    int t1 = NKERN * 24 * 32;  // 20736 threads
    int t2 = NKERN * 8 * 32;   // 6912 threads
    repack_kernel<<<(t1 + 255) / 256, 256, 0, stream>>>(br[b].W1, b1f[b], NKERN, 96, 24);
    repack_kernel<<<(t2 + 255) / 256, 256, 0, stream>>>(br[b].W2, b2f[b], NKERN, 32, 8);
  }

  int nTiles = (Nv + 15) / 16;
  int blocks = (nTiles + 7) / 8;  // 8 waves (tiles) per 256-thread block
  for (int b = 0; b < 3; ++b) {
    conv_stage<24, false><<<blocks, 256, 0, stream>>>(
        feats, nbr[b], b1f[b], br[b].b1, br[b].g1, br[b].be1, br[b].m1, br[b].v1,
        zerorow, nullptr, h, Nv, 0);
    conv_stage<8, true><<<blocks, 256, 0, stream>>>(
        h, nbr[b], b2f[b], br[b].b2, br[b].g2, br[b].be2, br[b].m2, br[b].v2,
        zerorow, feats, (float*)d_out, Nv, b * 32);
  }
}